// CausalSelfAttention_12146167513117
// MI455X (gfx1250) — compile-verified
//
#include <hip/hip_runtime.h>

typedef __attribute__((ext_vector_type(2))) float v2f;
typedef __attribute__((ext_vector_type(4))) float v4f;
typedef __attribute__((ext_vector_type(8))) float v8f;

#define WMMA_F32(a, b, c) \
  __builtin_amdgcn_wmma_f32_16x16x4_f32(false, (a), false, (b), (short)0, (c), false, false)

static __device__ __forceinline__ v8f v8zero() {
  v8f z = {0.f, 0.f, 0.f, 0.f, 0.f, 0.f, 0.f, 0.f};
  return z;
}

// Async 16-byte global -> LDS copy (CDNA5 GLOBAL_LOAD_ASYNC_TO_LDS_B128).
// LDS destination = low 32 bits of the generic shared-memory pointer
// (LDS aperture: addr[31:0] is the LDS offset). Tracked by ASYNCcnt.
static __device__ __forceinline__ void async_ld16(void* ldsDst, const void* gsrc) {
  unsigned l = (unsigned)(size_t)ldsDst;
  asm volatile("global_load_async_to_lds_b128 %0, %1, off"
               :: "v"(l), "v"(gsrc)
               : "memory");
}
static __device__ __forceinline__ void wait_async0() {
  asm volatile("s_wait_asynccnt 0" ::: "memory");
}

// ---------------------------------------------------------------------------
// Problem constants (from reference): B=2, T=2048, C=1024, H=16, d=64
// ---------------------------------------------------------------------------
static constexpr int Tn = 2048;
static constexpr int Cn = 1024;
static constexpr int Hn = 16;
static constexpr int Dn = 64;
static constexpr int Mn = 2 * Tn;   // 4096 rows (B*T)
static constexpr int N3 = 3 * Cn;   // 3072

// ---------------------------------------------------------------------------
// Kernel 1: qkv = x @ W_attn + b_attn, scattered into Q/K/V [B,H,T,64]
// (Q pre-scaled by 1/sqrt(d)). 256 thr / 8 waves, 128x128 tile, K-tile 16,
// double-buffered async LDS staging, wave = 64x32 (4x2 WMMA tiles).
// ---------------------------------------------------------------------------
__global__ __launch_bounds__(256)
void qkv_gemm_kernel(const float* __restrict__ x, const float* __restrict__ W,
                     const float* __restrict__ bias, float* __restrict__ Q,
                     float* __restrict__ K, float* __restrict__ V) {
  __shared__ float sA[2][128 * 20];   // 128 x 16, padded stride 20
  __shared__ float sB[2][16 * 128];

  const int tid  = threadIdx.x;
  const int wave = tid >> 5;
  const int lane = tid & 31;
  const int half = lane >> 4;
  const int l16  = lane & 15;
  const int m0   = blockIdx.y * 128;
  const int n0   = blockIdx.x * 128;
  const int mW   = (wave >> 2) * 64;
  const int nW   = (wave & 3) * 32;

  auto issue_tile = [&](int kt, int buf) {
#pragma unroll
    for (int i = 0; i < 2; ++i) {   // A: 512 x 16B
      int f4 = tid + i * 256;
      int r = f4 >> 2, c4 = f4 & 3;
      async_ld16(&sA[buf][r * 20 + c4 * 4],
                 x + (size_t)(m0 + r) * Cn + kt + c4 * 4);
    }
#pragma unroll
    for (int i = 0; i < 2; ++i) {   // B: 512 x 16B
      int f4 = tid + i * 256;
      int r = f4 >> 5, c4 = f4 & 31;
      async_ld16(&sB[buf][r * 128 + c4 * 4],
                 W + (size_t)(kt + r) * N3 + n0 + c4 * 4);
    }
  };

  v8f acc[4][2];
#pragma unroll
  for (int i = 0; i < 4; ++i)
#pragma unroll
    for (int j = 0; j < 2; ++j) acc[i][j] = v8zero();

  issue_tile(0, 0);
  int cur = 0;
  for (int kt = 0; kt < Cn; kt += 16) {
    wait_async0();
    __syncthreads();                       // tile[cur] ready, buf[cur^1] free
    if (kt + 16 < Cn) issue_tile(kt + 16, cur ^ 1);
    const float* As = sA[cur];
    const float* Bs = sB[cur];
#pragma unroll
    for (int ks = 0; ks < 4; ++ks) {
      const int k0 = ks * 4 + 2 * half;
      v2f bf[2];
#pragma unroll
      for (int j = 0; j < 2; ++j) {
        bf[j].x = Bs[k0 * 128 + nW + j * 16 + l16];
        bf[j].y = Bs[(k0 + 1) * 128 + nW + j * 16 + l16];
      }
#pragma unroll
      for (int i = 0; i < 4; ++i) {
        const float* ap = As + (mW + i * 16 + l16) * 20 + k0;
        v2f af;
        af.x = ap[0];
        af.y = ap[1];
#pragma unroll
        for (int j = 0; j < 2; ++j) acc[i][j] = WMMA_F32(af, bf[j], acc[i][j]);
      }
    }
    cur ^= 1;
  }

  // Epilogue: bias, head-major scatter, fold softmax scale into Q.
#pragma unroll
  for (int i = 0; i < 4; ++i) {
#pragma unroll
    for (int j = 0; j < 2; ++j) {
      const int n     = n0 + nW + j * 16 + l16;
      const float bv  = bias[n];
      const int which = n >> 10;   // 0=q 1=k 2=v
      const int c     = n & (Cn - 1);
      const int h     = c >> 6;
      const int dd    = c & 63;
      float* dst      = (which == 0) ? Q : ((which == 1) ? K : V);
      const float scl = (which == 0) ? 0.125f : 1.0f;
#pragma unroll
      for (int v = 0; v < 8; ++v) {
        const int m  = m0 + mW + i * 16 + v + 8 * half;
        const int bb = m >> 11;
        const int t  = m & (Tn - 1);
        dst[(((size_t)(bb * Hn + h)) * Tn + t) * Dn + dd] = (acc[i][j][v] + bv) * scl;
      }
    }
  }
}

// ---------------------------------------------------------------------------
// Kernel 2: causal flash attention per (b,h). Block = 128 query rows, 8 waves;
// each wave owns 16 rows. K/V streamed via double-buffered async LDS tiles.
// ---------------------------------------------------------------------------
__global__ __launch_bounds__(256)
void attn_kernel(const float* __restrict__ Q, const float* __restrict__ Kv,
                 const float* __restrict__ V, const float* __restrict__ amask,
                 float* __restrict__ Y) {
  __shared__ float sK[2][32 * 68];    // 32 keys x 64 dims, padded stride 68
  __shared__ float sV[2][32 * 64];
  __shared__ float sP[8][16 * 36];    // per-wave P scratch, padded stride 36

  const int tid   = threadIdx.x;
  const int wave  = tid >> 5;
  const int lane  = tid & 31;
  const int half  = lane >> 4;
  const int l16   = lane & 15;
  const int bh    = blockIdx.y;
  const int batch = bh >> 4;
  const int head  = bh & 15;
  const int qRow0 = blockIdx.x * 128 + wave * 16;

  const float* Qb = Q + (size_t)bh * Tn * Dn;
  const float* Kb = Kv + (size_t)bh * Tn * Dn;
  const float* Vb = V + (size_t)bh * Tn * Dn;

  auto issue_kv = [&](int kt, int buf) {
#pragma unroll
    for (int i = 0; i < 2; ++i) {
      int f4 = tid + i * 256;
      int r = f4 >> 4, c4 = f4 & 15;
      async_ld16(&sK[buf][r * 68 + c4 * 4],
                 Kb + (size_t)(kt * 32 + r) * Dn + c4 * 4);
      async_ld16(&sV[buf][r * 64 + c4 * 4],
                 Vb + (size_t)(kt * 32 + r) * Dn + c4 * 4);
    }
  };

  // Q tile (16x64) held in registers as A-fragments.
  v2f qa[16];
  {
    const float* qrow = Qb + (size_t)(qRow0 + l16) * Dn;
#pragma unroll
    for (int ks = 0; ks < 16; ++ks) {
      const int k = ks * 4 + 2 * half;
      qa[ks].x = qrow[k];
      qa[ks].y = qrow[k + 1];
    }
  }

  v8f acc_o[4];
#pragma unroll
  for (int jj = 0; jj < 4; ++jj) acc_o[jj] = v8zero();
  float m_run[8], l_run[8];
#pragma unroll
  for (int v = 0; v < 8; ++v) { m_run[v] = -3.0e38f; l_run[v] = 0.0f; }

  const int nTiles = blockIdx.x * 4 + 4;   // causal pruning
  issue_kv(0, 0);
  int cur = 0;
  for (int kt = 0; kt < nTiles; ++kt) {
    wait_async0();
    __syncthreads();                        // K/V[cur] ready, buf[cur^1] free
    if (kt + 1 < nTiles) issue_kv(kt + 1, cur ^ 1);
    const int kBase = kt * 32;
    const float* Ks = sK[cur];
    const float* Vs = sV[cur];

    // S = q @ K^T  (16 x 32)
    v8f s[2];
    s[0] = v8zero();
    s[1] = v8zero();
#pragma unroll
    for (int ks = 0; ks < 16; ++ks) {
      const int k = ks * 4 + 2 * half;
#pragma unroll
      for (int j = 0; j < 2; ++j) {
        const float* kp = Ks + (j * 16 + l16) * 68 + k;
        v2f bf;
        bf.x = kp[0];
        bf.y = kp[1];
        s[j] = WMMA_F32(qa[ks], bf, s[j]);
      }
    }

    // Causal mask + additive attention mask (reference order).
    float am[2];
#pragma unroll
    for (int j = 0; j < 2; ++j) am[j] = amask[batch * Tn + kBase + j * 16 + l16];
#pragma unroll
    for (int j = 0; j < 2; ++j) {
      const int col = kBase + j * 16 + l16;
#pragma unroll
      for (int v = 0; v < 8; ++v) {
        const int row = qRow0 + v + 8 * half;
        float sv = s[j][v];
        sv = (col <= row) ? sv : -10000.0f;
        s[j][v] = sv + am[j];
      }
    }

    // Online softmax (row = v + 8*half lives in one 16-lane half).
#pragma unroll
    for (int v = 0; v < 8; ++v) {
      float tmax = fmaxf(s[0][v], s[1][v]);
#pragma unroll
      for (int off = 1; off < 16; off <<= 1)
        tmax = fmaxf(tmax, __shfl_xor(tmax, off, 32));
      const float m_new = fmaxf(m_run[v], tmax);
      const float corr  = __expf(m_run[v] - m_new);
      const float p0 = __expf(s[0][v] - m_new);
      const float p1 = __expf(s[1][v] - m_new);
      s[0][v] = p0;
      s[1][v] = p1;
      float tsum = p0 + p1;
#pragma unroll
      for (int off = 1; off < 16; off <<= 1) tsum += __shfl_xor(tsum, off, 32);
      l_run[v] = l_run[v] * corr + tsum;
      m_run[v] = m_new;
#pragma unroll
      for (int jj = 0; jj < 4; ++jj) acc_o[jj][v] *= corr;
    }

    // C-layout -> A-layout for P via wave-private LDS bounce.
    float* pw = &sP[wave][0];
#pragma unroll
    for (int j = 0; j < 2; ++j)
#pragma unroll
      for (int v = 0; v < 8; ++v)
        pw[(v + 8 * half) * 36 + j * 16 + l16] = s[j][v];
    v2f pa[8];
#pragma unroll
    for (int ks = 0; ks < 8; ++ks) {
      const float* pp = pw + l16 * 36 + ks * 4 + 2 * half;
      pa[ks].x = pp[0];
      pa[ks].y = pp[1];
    }

    // O += P @ V
#pragma unroll
    for (int ks = 0; ks < 8; ++ks) {
      const int k = ks * 4 + 2 * half;
#pragma unroll
      for (int jj = 0; jj < 4; ++jj) {
        v2f bf;
        bf.x = Vs[k * 64 + jj * 16 + l16];
        bf.y = Vs[(k + 1) * 64 + jj * 16 + l16];
        acc_o[jj] = WMMA_F32(pa[ks], bf, acc_o[jj]);
      }
    }
    cur ^= 1;
  }

  // Finalize: O / l, write y to [B,T,C] (merges head transpose).
#pragma unroll
  for (int v = 0; v < 8; ++v) {
    const float inv = 1.0f / l_run[v];
    const int t = qRow0 + v + 8 * half;
#pragma unroll
    for (int jj = 0; jj < 4; ++jj) {
      const int c = head * Dn + jj * 16 + l16;
      Y[((size_t)batch * Tn + t) * Cn + c] = acc_o[jj][v] * inv;
    }
  }
}

// ---------------------------------------------------------------------------
// Kernel 3: out = y @ W_proj + b_proj   (4096 x 1024 x 1024)
// ---------------------------------------------------------------------------
__global__ __launch_bounds__(256)
void proj_gemm_kernel(const float* __restrict__ Yin, const float* __restrict__ W,
                      const float* __restrict__ bias, float* __restrict__ out) {
  __shared__ float sA[2][128 * 20];
  __shared__ float sB[2][16 * 128];

  const int tid  = threadIdx.x;
  const int wave = tid >> 5;
  const int lane = tid & 31;
  const int half = lane >> 4;
  const int l16  = lane & 15;
  const int m0   = blockIdx.y * 128;
  const int n0   = blockIdx.x * 128;
  const int mW   = (wave >> 2) * 64;
  const int nW   = (wave & 3) * 32;

  auto issue_tile = [&](int kt, int buf) {
#pragma unroll
    for (int i = 0; i < 2; ++i) {
      int f4 = tid + i * 256;
      int r = f4 >> 2, c4 = f4 & 3;
      async_ld16(&sA[buf][r * 20 + c4 * 4],
                 Yin + (size_t)(m0 + r) * Cn + kt + c4 * 4);
    }
#pragma unroll
    for (int i = 0; i < 2; ++i) {
      int f4 = tid + i * 256;
      int r = f4 >> 5, c4 = f4 & 31;
      async_ld16(&sB[buf][r * 128 + c4 * 4],
                 W + (size_t)(kt + r) * Cn + n0 + c4 * 4);
    }
  };

  v8f acc[4][2];
#pragma unroll
  for (int i = 0; i < 4; ++i)
#pragma unroll
    for (int j = 0; j < 2; ++j) acc[i][j] = v8zero();

  issue_tile(0, 0);
  int cur = 0;
  for (int kt = 0; kt < Cn; kt += 16) {
    wait_async0();
    __syncthreads();
    if (kt + 16 < Cn) issue_tile(kt + 16, cur ^ 1);
    const float* As = sA[cur];
    const float* Bs = sB[cur];
#pragma unroll
    for (int ks = 0; ks < 4; ++ks) {
      const int k0 = ks * 4 + 2 * half;
      v2f bf[2];
#pragma unroll
      for (int j = 0; j < 2; ++j) {
        bf[j].x = Bs[k0 * 128 + nW + j * 16 + l16];
        bf[j].y = Bs[(k0 + 1) * 128 + nW + j * 16 + l16];
      }
#pragma unroll
      for (int i = 0; i < 4; ++i) {
        const float* ap = As + (mW + i * 16 + l16) * 20 + k0;
        v2f af;
        af.x = ap[0];
        af.y = ap[1];
#pragma unroll
        for (int j = 0; j < 2; ++j) acc[i][j] = WMMA_F32(af, bf[j], acc[i][j]);
      }
    }
    cur ^= 1;
  }

#pragma unroll
  for (int i = 0; i < 4; ++i) {
#pragma unroll
    for (int j = 0; j < 2; ++j) {
      const int n    = n0 + nW + j * 16 + l16;
      const float bv = bias[n];
#pragma unroll
      for (int v = 0; v < 8; ++v) {
        const int m = m0 + mW + i * 16 + v + 8 * half;
        out[(size_t)m * Cn + n] = acc[i][j][v] + bv;
      }
    }
  }
}

// ---------------------------------------------------------------------------
extern "C" void kernel_launch(void* const* d_in, const int* in_sizes, int n_in,
                              void* d_out, int out_size, void* d_ws, size_t ws_size,
                              hipStream_t stream) {
  (void)in_sizes; (void)n_in; (void)out_size; (void)ws_size;

  const float* x      = (const float*)d_in[0];
  const float* amask  = (const float*)d_in[1];
  const float* W_attn = (const float*)d_in[2];
  const float* b_attn = (const float*)d_in[3];
  const float* W_proj = (const float*)d_in[4];
  const float* b_proj = (const float*)d_in[5];
  float* out = (float*)d_out;

  const size_t perBuf = (size_t)Mn * Cn;   // 16 MB each
  float* Q = (float*)d_ws;
  float* K = Q + perBuf;
  float* V = K + perBuf;
  float* Y = V + perBuf;

  dim3 g1(N3 / 128, Mn / 128);   // 24 x 32
  qkv_gemm_kernel<<<g1, 256, 0, stream>>>(x, W_attn, b_attn, Q, K, V);

  dim3 g2(Tn / 128, 2 * Hn);     // 16 x 32
  attn_kernel<<<g2, 256, 0, stream>>>(Q, K, V, amask, Y);

  dim3 g3(Cn / 128, Mn / 128);   // 8 x 32
  proj_gemm_kernel<<<g3, 256, 0, stream>>>(Y, W_proj, b_proj, out);
}